// GCNLayer_34273839022909
// MI455X (gfx1250) — compile-verified
//
#include <hip/hip_runtime.h>

#define N_NODES 20000
#define N_EDGES 640000
#define D 128

typedef __attribute__((ext_vector_type(2))) float v2f;
typedef __attribute__((ext_vector_type(8))) float v8f;

// ---------------------------------------------------------------------------
// Kernel 1: zero the aggregation workspace
// ---------------------------------------------------------------------------
__global__ void gcn_zero_kernel(float* __restrict__ p, int n) {
    int i = blockIdx.x * blockDim.x + threadIdx.x;
    if (i < n) p[i] = 0.0f;
}

// ---------------------------------------------------------------------------
// Kernel 2: scatter-add edge messages. One wave32 per edge; lane L handles
// features [4L, 4L+3] (wave reads 512 contiguous bytes of h[src], then 4
// f32 atomics into agg[dst]). agg (10.2 MB) lives in L2 (192 MB) so the
// non-returning global_atomic_add_f32 ops resolve at L2.
// ---------------------------------------------------------------------------
__global__ void __launch_bounds__(256)
gcn_scatter_kernel(const float* __restrict__ h,
                   const long long* __restrict__ eidx,
                   float* __restrict__ agg) {
    int wave = (blockIdx.x * blockDim.x + threadIdx.x) >> 5;   // edge id
    int lane = threadIdx.x & 31;
    if (wave >= N_EDGES) return;
    long long src = eidx[wave];              // edge_index[0][e]
    long long dst = eidx[N_EDGES + wave];    // edge_index[1][e]
    const float4 v = *((const float4*)(h + src * D) + lane);
    float* ap = agg + dst * D + lane * 4;
    unsafeAtomicAdd(ap + 0, v.x);
    unsafeAtomicAdd(ap + 1, v.y);
    unsafeAtomicAdd(ap + 2, v.z);
    unsafeAtomicAdd(ap + 3, v.w);
}

// ---------------------------------------------------------------------------
// Kernel 3: agg[i] *= 1 / max(deg[node], 1)
// ---------------------------------------------------------------------------
__global__ void gcn_scale_kernel(float* __restrict__ agg,
                                 const float* __restrict__ deg, int n) {
    int i = blockIdx.x * blockDim.x + threadIdx.x;
    if (i >= n) return;
    float d = deg[i >> 7];                 // i / D
    float r = 1.0f / fmaxf(d, 1.0f);
    agg[i] *= r;
}

// ---------------------------------------------------------------------------
// Kernel 4: fused dual-GEMM + bias + ReLU with V_WMMA_F32_16X16X4_F32.
//   out = relu(h @ Ws^T + b + aggs @ Wn^T)
// One block = one 16-row node tile. 8 waves = 8 column tiles of 16.
// Each wave: 32 K-steps over h/Ws then 32 over aggs/Wn, one shared v8f acc.
//
// A (16x4 f32, 2 VGPRs/lane): lane<16 -> {A[M=lane,k0], A[M=lane,k0+1]},
//                             lane>=16 -> {A[M=lane-16,k0+2], A[...,k0+3]}
// B (4x16 f32, 2 VGPRs/lane): lane gives {B[k0+2*hi][n], B[k0+2*hi+1][n]},
//                             n = lane&15; B[k][n] = W[n][k]  (x @ W.T)
// C/D (16x16 f32, 8 VGPRs): VGPR r -> M = r + 8*hi, N = lane&15.
// ---------------------------------------------------------------------------
__global__ void __launch_bounds__(256)
gcn_gemm_kernel(const float* __restrict__ h,
                const float* __restrict__ aggs,
                const float* __restrict__ w_self,
                const float* __restrict__ b_self,
                const float* __restrict__ w_nei,
                float* __restrict__ out) {
    const int m0   = blockIdx.x * 16;        // node tile
    const int wv   = threadIdx.x >> 5;       // 0..7: output column tile
    const int lane = threadIdx.x & 31;
    const int n0   = wv * 16;
    const int r    = lane & 15;
    const int hi   = lane >> 4;              // 0 or 1

    v8f c = {};

    const float* arow  = h      + (long long)(m0 + r) * D;  // A row (M = m0+r)
    const float* brows = w_self + (long long)(n0 + r) * D;  // Ws row n = n0+r
    #pragma unroll 4
    for (int k0 = 0; k0 < D; k0 += 4) {
        const int ka = k0 + 2 * hi;
        v2f a = { arow[ka],  arow[ka + 1]  };
        v2f b = { brows[ka], brows[ka + 1] };
        c = __builtin_amdgcn_wmma_f32_16x16x4_f32(
                /*neg_a=*/false, a, /*neg_b=*/false, b,
                /*c_mod=*/(short)0, c, /*reuse_a=*/false, /*reuse_b=*/false);
    }

    const float* arow2 = aggs  + (long long)(m0 + r) * D;
    const float* brown = w_nei + (long long)(n0 + r) * D;
    #pragma unroll 4
    for (int k0 = 0; k0 < D; k0 += 4) {
        const int ka = k0 + 2 * hi;
        v2f a = { arow2[ka], arow2[ka + 1] };
        v2f b = { brown[ka], brown[ka + 1] };
        c = __builtin_amdgcn_wmma_f32_16x16x4_f32(
                false, a, false, b, (short)0, c, false, false);
    }

    const float bias  = b_self[n0 + r];
    const int   mbase = m0 + hi * 8;
    #pragma unroll
    for (int i = 0; i < 8; ++i) {
        float v = c[i] + bias;
        v = v > 0.0f ? v : 0.0f;
        out[(long long)(mbase + i) * D + n0 + r] = v;
    }
}

// ---------------------------------------------------------------------------
extern "C" void kernel_launch(void* const* d_in, const int* in_sizes, int n_in,
                              void* d_out, int out_size, void* d_ws, size_t ws_size,
                              hipStream_t stream) {
    const float*     h      = (const float*)d_in[0];
    const long long* eidx   = (const long long*)d_in[1];   // int64 [2, E]
    const float*     deg    = (const float*)d_in[2];
    const float*     w_self = (const float*)d_in[3];
    const float*     b_self = (const float*)d_in[4];
    const float*     w_nei  = (const float*)d_in[5];
    float*           out    = (float*)d_out;
    float*           agg    = (float*)d_ws;                // N_NODES*D floats

    const int nelem = N_NODES * D;                         // 2,560,000

    // 1) agg = 0
    gcn_zero_kernel<<<(nelem + 255) / 256, 256, 0, stream>>>(agg, nelem);

    // 2) scatter-add: one wave per edge, 8 edges per 256-thread block
    gcn_scatter_kernel<<<N_EDGES / 8, 256, 0, stream>>>(h, eidx, agg);

    // 3) agg /= max(deg, 1)
    gcn_scale_kernel<<<(nelem + 255) / 256, 256, 0, stream>>>(agg, deg, nelem);

    // 4) fused dual GEMM + bias + relu (WMMA f32)
    gcn_gemm_kernel<<<N_NODES / 16, 256, 0, stream>>>(h, agg, w_self, b_self,
                                                      w_nei, out);
}